// BasicRoIExtractor_6098853560989
// MI455X (gfx1250) — compile-verified
//
#include <hip/hip_runtime.h>

#define OUT_SZ 7
#define NSAMP 14            // OUT_SZ * RATIO sample points per dim
#define CBLK 8              // channels staged per workgroup
#define AREA_MAX 1312       // provable bound on region area
#define CSTR 1313           // LDS per-channel stride (odd -> bank spread)
#define NTHREADS 256
#define NWAVES (NTHREADS / 32)

typedef __attribute__((address_space(1))) int* gptr_i32;
typedef __attribute__((address_space(3))) int* lptr_i32;

// ---- CDNA5 async global->LDS copy (guarded; sync fallback keeps compile green)
#if defined(__gfx1250__) && __has_builtin(__builtin_amdgcn_global_load_async_to_lds_b32)
#define HAVE_ASYNC_LDS 1
__device__ __forceinline__ void async_cp_b32(const float* g, float* l) {
  float* gnc = const_cast<float*>(g);
  __builtin_amdgcn_global_load_async_to_lds_b32(
      (gptr_i32)gnc, (lptr_i32)l, 0, 0);
}
__device__ __forceinline__ void async_wait0() {
#if __has_builtin(__builtin_amdgcn_s_wait_asynccnt)
  __builtin_amdgcn_s_wait_asynccnt(0);
#else
  asm volatile("s_wait_asynccnt 0" ::: "memory");
#endif
}
#else
#define HAVE_ASYNC_LDS 0
__device__ __forceinline__ void async_cp_b32(const float* g, float* l) { *l = *g; }
__device__ __forceinline__ void async_wait0() {}
#endif

__global__ __launch_bounds__(NTHREADS)
void roi_extractor_kernel(const float* __restrict__ f0,
                          const float* __restrict__ f1,
                          const float* __restrict__ f2,
                          const float* __restrict__ f3,
                          const float* __restrict__ rois,
                          float* __restrict__ out, int K) {
  __shared__ float s_plane[CBLK * CSTR];
  __shared__ float s_wy[NSAMP], s_vy[NSAMP], s_wx[NSAMP], s_vx[NSAMP];
  __shared__ int   s_y0[NSAMP], s_x0[NSAMP];
  __shared__ int   s_par[6];   // ylo, xlo, Hr, Wr, area, ok

  const int k   = blockIdx.x;
  const int cb  = blockIdx.y;
  const int tid = threadIdx.x;

  // ---- RoI params (wave-uniform; compiler scalarizes to s_*_f32)
  const float x1 = rois[0 * K + k], y1 = rois[1 * K + k];
  const float x2 = rois[2 * K + k], y2 = rois[3 * K + k];
  const float sz = sqrtf((x2 - x1 + 1.0f) * (y2 - y1 + 1.0f));
  int lv = (int)floorf(log2f(sz / 56.0f + 1e-6f));
  lv = lv < 0 ? 0 : (lv > 3 ? 3 : lv);

  const float* feat = f0; int H = 200, W = 304; float scale = 0.25f;
  if (lv == 1) { feat = f1; H = 100; W = 152; scale = 0.125f; }
  else if (lv == 2) { feat = f2; H = 50; W = 76; scale = 0.0625f; }
  else if (lv == 3) { feat = f3; H = 25; W = 38; scale = 0.03125f; }

  const float rx1 = x1 * scale, ry1 = y1 * scale;
  const float rw = fmaxf(x2 * scale - rx1, 1.0f);
  const float rh = fmaxf(y2 * scale - ry1, 1.0f);

  // ---- sample coordinates / weights / validity into LDS
  if (tid < NSAMP) {
    const float g = (tid + 0.5f) * 0.5f;
    const float yc = ry1 + g * (rh * (1.0f / OUT_SZ));
    const float v = (yc >= -1.0f && yc <= (float)H) ? 1.0f : 0.0f;
    const float y = fminf(fmaxf(yc, 0.0f), (float)(H - 1));
    const float yf = floorf(y);
    s_y0[tid] = (int)yf; s_wy[tid] = y - yf; s_vy[tid] = v;
  } else if (tid < 2 * NSAMP) {
    const int i = tid - NSAMP;
    const float g = (i + 0.5f) * 0.5f;
    const float xc = rx1 + g * (rw * (1.0f / OUT_SZ));
    const float v = (xc >= -1.0f && xc <= (float)W) ? 1.0f : 0.0f;
    const float x = fminf(fmaxf(xc, 0.0f), (float)(W - 1));
    const float xf = floorf(x);
    s_x0[i] = (int)xf; s_wx[i] = x - xf; s_vx[i] = v;
  }
  __syncthreads();

  if (tid == 0) {
    const int ylo = s_y0[0], yhi = min(s_y0[NSAMP - 1] + 1, H - 1);
    const int xlo = s_x0[0], xhi = min(s_x0[NSAMP - 1] + 1, W - 1);
    const int Hr = yhi - ylo + 1, Wr = xhi - xlo + 1;
    const int area = Hr * Wr;
    s_par[0] = ylo; s_par[1] = xlo; s_par[2] = Hr; s_par[3] = Wr;
    s_par[4] = area; s_par[5] = (area <= AREA_MAX) ? 1 : 0;
  }
  __syncthreads();

  const int ylo = s_par[0], xlo = s_par[1], Hr = s_par[2], Wr = s_par[3];
  const int ok = s_par[5];
  const int HW = H * W;
  const float* fbase = feat + (size_t)(cb * CBLK) * HW;

  // ---- stage region tile [CBLK x Hr x Wr] into LDS via async copies.
  // wave-per-row, lane-per-column: lanes issue contiguous B32 async copies
  // (coalesced into 128B lines), no integer division.
  if (ok) {
    const int lane = tid & 31;
    const int wv   = tid >> 5;
    for (int c = 0; c < CBLK; ++c) {
      const float* src = fbase + (size_t)c * HW + (size_t)ylo * W + xlo;
      float* dst = s_plane + c * CSTR;
      for (int r = wv; r < Hr; r += NWAVES) {
        const float* srow = src + (size_t)r * W;
        float* drow = dst + r * Wr;
        for (int q = lane; q < Wr; q += 32) {
          async_cp_b32(srow + q, drow + q);
        }
      }
    }
    async_wait0();
  }
  __syncthreads();

  // ---- compute: each thread handles (channel, bin) pairs; stores contiguous
  if (ok) {
    for (int t = tid; t < CBLK * 49; t += NTHREADS) {
      const int c = t / 49, bin = t - c * 49;
      const int by = bin / 7, bx = bin - by * 7;
      const float* p = s_plane + c * CSTR;
      float acc = 0.0f;
#pragma unroll
      for (int i = 0; i < 2; ++i) {
        const int sy = 2 * by + i;
        const int yy0 = s_y0[sy];
        const int r0 = (yy0 - ylo) * Wr;
        const int r1 = (min(yy0 + 1, H - 1) - ylo) * Wr;
        const float wy = s_wy[sy], vy = s_vy[sy];
        const float omwy = 1.0f - wy;
#pragma unroll
        for (int j = 0; j < 2; ++j) {
          const int sx = 2 * bx + j;
          const int q0 = s_x0[sx] - xlo;
          const int q1 = min(s_x0[sx] + 1, W - 1) - xlo;
          const float wx = s_wx[sx];
          const float omwx = 1.0f - wx;
          const float vv = vy * s_vx[sx];
          const float v00 = p[r0 + q0], v01 = p[r0 + q1];
          const float v10 = p[r1 + q0], v11 = p[r1 + q1];
          acc += vv * (v00 * omwy * omwx + v01 * omwy * wx +
                       v10 * wy * omwx + v11 * wy * wx);
        }
      }
      out[((size_t)k * 256 + (size_t)cb * CBLK + c) * 49 + bin] = acc * 0.25f;
    }
  } else {  // safety fallback: direct global gather (region larger than LDS tile)
    for (int t = tid; t < CBLK * 49; t += NTHREADS) {
      const int c = t / 49, bin = t - c * 49;
      const int by = bin / 7, bx = bin - by * 7;
      const float* p = fbase + (size_t)c * HW;
      float acc = 0.0f;
#pragma unroll
      for (int i = 0; i < 2; ++i) {
        const int sy = 2 * by + i;
        const int yy0 = s_y0[sy];
        const int yy1 = min(yy0 + 1, H - 1);
        const float wy = s_wy[sy], vy = s_vy[sy];
        const float omwy = 1.0f - wy;
#pragma unroll
        for (int j = 0; j < 2; ++j) {
          const int sx = 2 * bx + j;
          const int xx0 = s_x0[sx];
          const int xx1 = min(xx0 + 1, W - 1);
          const float wx = s_wx[sx];
          const float omwx = 1.0f - wx;
          const float vv = vy * s_vx[sx];
          const float v00 = p[(size_t)yy0 * W + xx0], v01 = p[(size_t)yy0 * W + xx1];
          const float v10 = p[(size_t)yy1 * W + xx0], v11 = p[(size_t)yy1 * W + xx1];
          acc += vv * (v00 * omwy * omwx + v01 * omwy * wx +
                       v10 * wy * omwx + v11 * wy * wx);
        }
      }
      out[((size_t)k * 256 + (size_t)cb * CBLK + c) * 49 + bin] = acc * 0.25f;
    }
  }
}

extern "C" void kernel_launch(void* const* d_in, const int* in_sizes, int n_in,
                              void* d_out, int out_size, void* d_ws, size_t ws_size,
                              hipStream_t stream) {
  const float* f0 = (const float*)d_in[0];
  const float* f1 = (const float*)d_in[1];
  const float* f2 = (const float*)d_in[2];
  const float* f3 = (const float*)d_in[3];
  const float* rois = (const float*)d_in[4];
  const int K = in_sizes[4] / 4;
  dim3 grid((unsigned)K, 256 / CBLK, 1);
  roi_extractor_kernel<<<grid, NTHREADS, 0, stream>>>(
      f0, f1, f2, f3, rois, (float*)d_out, K);
}